// DecodeYoloV2_46153718563299
// MI455X (gfx1250) — compile-verified
//
#include <hip/hip_runtime.h>
#include <stdint.h>

#define NUM_CLASSES 80
#define A_NUM       5
#define GRID_W      208
#define HW          43264       /* 208*208 */
#define N_DET       216320      /* 5*HW */
#define CH_PER_A    85
#define STRIDE_F    32.0f       /* 6656/208 */
#define CONF_THR    0.5f
#define IOU_THR     0.5f
#define TOPK        1024
#define NBINS       4096
#define TILE_S      128         /* spatial positions per decode tile */
#define TILES_PER_A 338         /* HW / TILE_S (even -> pairs never straddle anchors) */
#define DECODE_BLOCKS (A_NUM * TILES_PER_A / 2)   /* 845 blocks, 2 tiles each */
#define TILE_ELEMS  (CH_PER_A * TILE_S)

/* workspace layout (bytes) */
#define OFF_BOXES   0u           /* N_DET * float4  = 3,461,120 */
#define OFF_SCORE   3461120u     /* N_DET * f32     =   865,280 */
#define OFF_CLS     4326400u     /* N_DET * i32     =   865,280 */
#define OFF_HIST    5191680u     /* 4096  * i32     =    16,384 */
#define OFF_CTRS    5208064u     /* 8     * i32                  */
#define OFF_TKS     5208096u     /* 1024  * f32                  */
#define OFF_TKI     5212192u     /* 1024  * i32                  */
#define OFF_TKB     5216288u     /* 1024  * float4  =    16,384  */
#define OFF_TKC     5232672u     /* 1024  * i32                  */
#define OFF_MASK    5236768u     /* 1024*32 * u32   =   131,072  */
#define WS_NEEDED   5367840u

typedef unsigned int u32x4 __attribute__((ext_vector_type(4)));
typedef int          i32x4 __attribute__((ext_vector_type(4)));
typedef int          i32x8 __attribute__((ext_vector_type(8)));

#if defined(__has_builtin)
#if __has_builtin(__builtin_amdgcn_tensor_load_to_lds) && \
    __has_builtin(__builtin_amdgcn_s_wait_tensorcnt)
#define HAVE_TDM 1
#endif
#endif

__device__ __forceinline__ float sigmoidf_(float x) {
    return 1.0f / (1.0f + __expf(-x));
}

#if defined(HAVE_TDM)
/* Issue one TDM DMA: 85 rows x 128 f32 cols, row stride 43264 elems, into LDS. */
__device__ __forceinline__ void tdm_issue_85x128(const float* gbase, const float* ldsdst) {
    const uint64_t ga  = (uint64_t)(uintptr_t)gbase;
    const uint32_t lds = (uint32_t)(uintptr_t)ldsdst;
    u32x4 g0;
    g0.x = 1u;                                               /* count=1 valid D# */
    g0.y = lds;                                              /* lds_addr */
    g0.z = (uint32_t)ga;                                     /* global_addr[31:0] */
    g0.w = (uint32_t)((ga >> 32) & 0x1FFFFFFu) | (2u << 30); /* addr[56:32] | type=2 */
    i32x8 g1;
    g1[0] = 0x00020000;                                      /* data_size = 4B */
    g1[1] = (int)(((unsigned)HW & 0xFFFFu) << 16);           /* tensor_dim0 lo16 */
    g1[2] = (int)((((unsigned)HW >> 16) & 0xFFFFu) | (425u << 16)); /* dim0 hi | dim1 lo */
    g1[3] = (int)((unsigned)TILE_S << 16);                   /* dim1 hi=0 | tile_dim0=128 */
    g1[4] = CH_PER_A;                                        /* tile_dim1=85, tile_dim2=0 */
    g1[5] = HW;                                              /* tensor_dim0_stride lo32 */
    g1[6] = 0;
    g1[7] = 0;
    i32x4 gz = {0, 0, 0, 0};
#if __clang_major__ >= 23
    i32x8 gz8 = {0, 0, 0, 0, 0, 0, 0, 0};
    __builtin_amdgcn_tensor_load_to_lds(g0, g1, gz, gz, gz8, 0);
#else
    __builtin_amdgcn_tensor_load_to_lds(g0, g1, gz, gz, 0);
#endif
}
#endif

/* ---- zero per-call scratch state (graph-replay safe) ---- */
__global__ __launch_bounds__(1024) void k_init(int* __restrict__ hist,
                                               int* __restrict__ ctrs,
                                               float* __restrict__ tks,
                                               int* __restrict__ tki) {
    const int t = threadIdx.x;
#pragma unroll
    for (int k = 0; k < NBINS / 1024; ++k) hist[t + k * 1024] = 0;
    if (t < 8) ctrs[t] = 0;
    tks[t] = -1.0f;
    tki[t] = 0;
}

/* ---- per-tile decode: box math + conf + 80-class argmax ---- */
__device__ __forceinline__ void decode_one(const float* __restrict__ tile,
                                           int a, int s0,
                                           const float* __restrict__ anchors,
                                           float4* __restrict__ boxes,
                                           float* __restrict__ score,
                                           int* __restrict__ cls,
                                           int* __restrict__ hist) {
    const int   t  = threadIdx.x;
    const int   s  = s0 + t;
    const float tx = tile[0 * TILE_S + t];
    const float ty = tile[1 * TILE_S + t];
    const float tw = tile[2 * TILE_S + t];
    const float th = tile[3 * TILE_S + t];
    const float tc = tile[4 * TILE_S + t];
    const float ax = anchors[2 * a];
    const float ay = anchors[2 * a + 1];
    const int wi = s % GRID_W;
    const int hi = s / GRID_W;
    float4 bb;
    bb.x = (sigmoidf_(tx) + (float)wi) * STRIDE_F;
    bb.y = (sigmoidf_(ty) + (float)hi) * STRIDE_F;
    bb.z = __expf(tw) * ax * STRIDE_F;
    bb.w = __expf(th) * ay * STRIDE_F;
    const float conf = sigmoidf_(tc);

    float m  = tile[5 * TILE_S + t];
    int   mi = 0;
#pragma unroll 8
    for (int c = 1; c < NUM_CLASSES; ++c) {
        const float v = tile[(5 + c) * TILE_S + t];
        if (v > m) { m = v; mi = c; }     /* first-max, matches jnp.argmax */
    }

    const int d = a * HW + s;
    boxes[d] = bb;
    cls[d]   = mi;
    if (conf > CONF_THR) {
        score[d] = conf;
        const uint32_t bits = __float_as_uint(conf);  /* conf in (0.5,1): bits in [0x3F000000,0x3F800000) */
        int bin = (int)((bits - 0x3F000000u) >> 11);
        bin = bin < 0 ? 0 : (bin > NBINS - 1 ? NBINS - 1 : bin);
        atomicAdd(&hist[bin], 1);
    } else {
        score[d] = -1.0f;
    }
}

/* ---- decode: double-buffered TDM pipeline, 2 tiles per block ---- */
__global__ __launch_bounds__(TILE_S) void k_decode(const float* __restrict__ x,
                                                   const float* __restrict__ anchors,
                                                   float4* __restrict__ boxes,
                                                   float* __restrict__ score,
                                                   int* __restrict__ cls,
                                                   int* __restrict__ hist) {
    __shared__ float tile[2 * TILE_ELEMS];     /* 87,040 B */
    const int pair = blockIdx.x;               /* 845 tile pairs */
    const int t0   = 2 * pair;
    const int a    = t0 / TILES_PER_A;         /* same anchor for both tiles */
    const int s00  = (t0 - a * TILES_PER_A) * TILE_S;
    const int s01  = s00 + TILE_S;
    const float* gb0 = x + (size_t)a * CH_PER_A * HW + s00;
    const float* gb1 = gb0 + TILE_S;

#if defined(HAVE_TDM)
    if (threadIdx.x < 32) {   /* wave 0 only: TDM ignores EXEC, one issue per wave */
        tdm_issue_85x128(gb0, &tile[0]);
        tdm_issue_85x128(gb1, &tile[TILE_ELEMS]);
        __builtin_amdgcn_s_wait_tensorcnt(1);   /* tile 0 done; tile 1 still in flight */
    }
#else
    for (int i = threadIdx.x; i < 2 * TILE_ELEMS; i += TILE_S) {
        const int buf = (i >= TILE_ELEMS) ? 1 : 0;
        const int r   = i - buf * TILE_ELEMS;
        const int c   = r >> 7;                /* / TILE_S */
        const int t2  = r & (TILE_S - 1);
        tile[i] = (buf ? gb1 : gb0)[(size_t)c * HW + t2];
    }
#endif
    __syncthreads();
    decode_one(&tile[0], a, s00, anchors, boxes, score, cls, hist);

#if defined(HAVE_TDM)
    if (threadIdx.x < 32) {
        __builtin_amdgcn_s_wait_tensorcnt(0);   /* tile 1 done */
    }
#endif
    __syncthreads();
    decode_one(&tile[TILE_ELEMS], a, s01, anchors, boxes, score, cls, hist);
}

/* ---- find threshold bin T for the top-K via block suffix scan ---- */
__global__ __launch_bounds__(1024) void k_thresh(const int* __restrict__ hist,
                                                 int* __restrict__ ctrs) {
    __shared__ int suf[1024];
    const int t  = threadIdx.x;
    const int h0 = hist[4 * t], h1 = hist[4 * t + 1], h2 = hist[4 * t + 2], h3 = hist[4 * t + 3];
    suf[t] = h0 + h1 + h2 + h3;
    __syncthreads();
    for (int off = 1; off < 1024; off <<= 1) {   /* inclusive suffix scan */
        const int xv = suf[t];
        const int yv = (t + off < 1024) ? suf[t + off] : 0;
        __syncthreads();
        suf[t] = xv + yv;
        __syncthreads();
    }
    const int base = (t < 1023) ? suf[t + 1] : 0;
    const int bs4 = base;
    const int bs3 = bs4 + h3;
    const int bs2 = bs3 + h2;
    const int bs1 = bs2 + h1;
    const int bs0 = bs1 + h0;
    const int bsa[5] = {bs0, bs1, bs2, bs3, bs4};   /* binSuffix for bins 4t..4t+3, and next */
#pragma unroll
    for (int k = 0; k < 4; ++k) {
        if (bsa[k] >= TOPK && bsa[k + 1] < TOPK) {
            ctrs[2] = 4 * t + k;     /* threshold bin T */
            ctrs[3] = bsa[k + 1];    /* #candidates strictly above bin T */
        }
    }
    if (t == 0 && bs0 < TOPK) {      /* fewer than TOPK candidates total */
        ctrs[2] = 0;
        ctrs[3] = bs1;
    }
}

/* ---- compact top-K candidates (unsorted) ---- */
__global__ __launch_bounds__(256) void k_compact(const float* __restrict__ score,
                                                 int* __restrict__ ctrs,
                                                 float* __restrict__ tks,
                                                 int* __restrict__ tki) {
    const int d = blockIdx.x * 256 + threadIdx.x;
    const float sc = score[d];
    if (!(sc > CONF_THR)) return;
    const uint32_t bits = __float_as_uint(sc);
    int bin = (int)((bits - 0x3F000000u) >> 11);
    bin = bin < 0 ? 0 : (bin > NBINS - 1 ? NBINS - 1 : bin);
    const int T  = ctrs[2];
    const int cA = ctrs[3];
    if (bin > T) {
        const int slot = atomicAdd(&ctrs[0], 1);
        if (slot < TOPK) { tks[slot] = sc; tki[slot] = d; }
    } else if (bin == T) {
        const int e    = atomicAdd(&ctrs[1], 1);
        const int slot = cA + e;
        if (slot < TOPK) { tks[slot] = sc; tki[slot] = d; }
    }
}

/* ---- bitonic sort 1024 (score desc, idx asc) + gather boxes/cls ---- */
__global__ __launch_bounds__(1024) void k_sort(float* __restrict__ tks,
                                               int* __restrict__ tki,
                                               const float4* __restrict__ boxes,
                                               const int* __restrict__ cls,
                                               float4* __restrict__ tkb,
                                               int* __restrict__ tkc) {
    __shared__ float s[TOPK];
    __shared__ int   id[TOPK];
    const int t = threadIdx.x;
    s[t]  = tks[t];
    id[t] = tki[t];
    __syncthreads();
    for (int k = 2; k <= TOPK; k <<= 1) {
        for (int j = k >> 1; j > 0; j >>= 1) {
            const int ixj = t ^ j;
            if (ixj > t) {
                const float av = s[t],  bv = s[ixj];
                const int   ai = id[t], bi = id[ixj];
                const bool before = (av > bv) || (av == bv && ai < bi);
                const bool asc    = (t & k) != 0;          /* overall descending */
                if (asc == before) { s[t] = bv; s[ixj] = av; id[t] = bi; id[ixj] = ai; }
            }
            __syncthreads();
        }
    }
    const float sv = s[t];
    const int   iv = id[t];
    tks[t] = sv;
    tkb[t] = boxes[iv];
    tkc[t] = cls[iv];
}

/* ---- 1024x1024 suppression bitmatrix (reference IoU formula) ---- */
__global__ __launch_bounds__(256) void k_iou(const float4* __restrict__ tkb,
                                             const int* __restrict__ tkc,
                                             const float* __restrict__ tks,
                                             unsigned* __restrict__ mask) {
    const int gid = blockIdx.x * 256 + threadIdx.x;   /* 32768 threads: (row, word) */
    const int i = gid >> 5;
    const int w = gid & 31;
    unsigned word = 0;
    const float si = tks[i];
    if (si > CONF_THR) {
        const float4 bi  = tkb[i];
        const int    ci  = tkc[i];
        const float xi0 = (bi.x - bi.z) * 0.5f;
        const float yi0 = (bi.y - bi.w) * 0.5f;
        const float xi1 = (bi.x + bi.z) * 0.5f;
        const float yi1 = (bi.y + bi.w) * 0.5f;
        const float a1  = fabsf((xi1 - xi0) * (yi1 - yi0));
#pragma unroll 4
        for (int r = 0; r < 32; ++r) {
            const int j = w * 32 + r;
            if (j <= i) continue;
            if (tkc[j] != ci) continue;
            const float4 bj  = tkb[j];
            const float xj0 = (bj.x - bj.z) * 0.5f;
            const float yj0 = (bj.y - bj.w) * 0.5f;
            const float xj1 = (bj.x + bj.z) * 0.5f;
            const float yj1 = (bj.y + bj.w) * 0.5f;
            const float iw = fminf(xi1, xj1) - fmaxf(xi0, xj0);
            const float ih = fminf(yi1, yj1) - fmaxf(yi0, yj0);
            const float inter = fmaxf(iw, 0.0f) * fmaxf(ih, 0.0f);
            const float a2 = fabsf((xj1 - xj0) * (yj1 - yj0));
            const float iou = inter / (a1 + a2 - inter + 1e-6f);
            if (iou >= IOU_THR) word |= (1u << r);
        }
    }
    mask[i * 32 + w] = word;
}

/* ---- greedy reduction (32-row LDS tiles, wave-serial chain) + output ---- */
__global__ __launch_bounds__(1024) void k_greedy(const float4* __restrict__ tkb,
                                                 const int* __restrict__ tkc,
                                                 const float* __restrict__ tks,
                                                 const unsigned* __restrict__ mask,
                                                 float* __restrict__ out) {
    __shared__ float    sc[TOPK];
    __shared__ unsigned dead[32];
    __shared__ unsigned tm[32 * 32];
    const int t = threadIdx.x;
    sc[t] = tks[t];
    if (t < 32) dead[t] = 0u;
    __syncthreads();
    const int r0 = t >> 5;
    const int w0 = t & 31;
    for (int tile = 0; tile < 32; ++tile) {
        tm[r0 * 32 + w0] = mask[(tile * 32 + r0) * 32 + w0];  /* parallel tile stage */
        __syncthreads();
        if (t < 32) {                                         /* wave 0: serial greedy, LDS only */
            unsigned my  = dead[t];
            unsigned cur = dead[tile];                        /* dead word covering this tile */
            for (int r = 0; r < 32; ++r) {
                const int i = tile * 32 + r;
                const bool alive = (sc[i] > CONF_THR) && (((cur >> r) & 1u) == 0u);
                if (alive) {
                    my  |= tm[r * 32 + t];
                    cur |= tm[r * 32 + tile];
                }
            }
            dead[t] = my;
        }
        __syncthreads();
    }
    const bool alive = (sc[t] > CONF_THR) && (((dead[t >> 5] >> (t & 31)) & 1u) == 0u);
    const float4 bb = tkb[t];
    const int    c  = tkc[t];
    float* o = out + t * 6;
    o[0] = alive ? bb.x : 0.0f;
    o[1] = alive ? bb.y : 0.0f;
    o[2] = alive ? bb.z : 0.0f;
    o[3] = alive ? bb.w : 0.0f;
    o[4] = alive ? sc[t] : 0.0f;
    o[5] = alive ? (float)c : 0.0f;
}

extern "C" void kernel_launch(void* const* d_in, const int* in_sizes, int n_in,
                              void* d_out, int out_size, void* d_ws, size_t ws_size,
                              hipStream_t stream) {
    (void)in_sizes; (void)n_in; (void)out_size;
    if (ws_size < WS_NEEDED) return;

    const float* x       = (const float*)d_in[0];
    const float* anchors = (const float*)d_in[1];
    char* ws = (char*)d_ws;
    float4*   boxes = (float4*)(ws + OFF_BOXES);
    float*    score = (float*)(ws + OFF_SCORE);
    int*      cls   = (int*)(ws + OFF_CLS);
    int*      hist  = (int*)(ws + OFF_HIST);
    int*      ctrs  = (int*)(ws + OFF_CTRS);
    float*    tks   = (float*)(ws + OFF_TKS);
    int*      tki   = (int*)(ws + OFF_TKI);
    float4*   tkb   = (float4*)(ws + OFF_TKB);
    int*      tkc   = (int*)(ws + OFF_TKC);
    unsigned* mask  = (unsigned*)(ws + OFF_MASK);
    float*    out   = (float*)d_out;

    k_init   <<<1, 1024, 0, stream>>>(hist, ctrs, tks, tki);
    k_decode <<<DECODE_BLOCKS, TILE_S, 0, stream>>>(x, anchors, boxes, score, cls, hist);
    k_thresh <<<1, 1024, 0, stream>>>(hist, ctrs);
    k_compact<<<N_DET / 256, 256, 0, stream>>>(score, ctrs, tks, tki);
    k_sort   <<<1, 1024, 0, stream>>>(tks, tki, boxes, cls, tkb, tkc);
    k_iou    <<<128, 256, 0, stream>>>(tkb, tkc, tks, mask);
    k_greedy <<<1, 1024, 0, stream>>>(tkb, tkc, tks, mask, out);
}